// GAT_79766132621995
// MI455X (gfx1250) — compile-verified
//
#include <hip/hip_runtime.h>
#include <hip/hip_bf16.h>
#include <math.h>
#include <stdint.h>

typedef __attribute__((ext_vector_type(2))) float v2f;
typedef __attribute__((ext_vector_type(8))) float v8f;

#define NEG_SLOPE 0.2f

// ---------------------------------------------------------------- utilities
__global__ void fill_kernel(float* __restrict__ p, long long n, float v) {
    long long i = (long long)blockIdx.x * blockDim.x + threadIdx.x;
    if (i < n) p[i] = v;
}

template <int F>
__global__ void bias_init_kernel(float* __restrict__ acc, const float* __restrict__ b,
                                 long long n) {
    long long i = (long long)blockIdx.x * blockDim.x + threadIdx.x;
    if (i < n) acc[i] = b[i % F];
}

__global__ void elu_kernel(float* __restrict__ x, long long n) {
    long long i = (long long)blockIdx.x * blockDim.x + threadIdx.x;
    if (i < n) {
        float v = x[i];
        x[i] = v > 0.f ? v : (expf(v) - 1.f);
    }
}

// ---------------------------------------------------------------- WMMA GEMM
// C[M,NN] = A[M,K] * B[K,NN], row-major fp32 via V_WMMA_F32_16X16X4_F32.
// One wave per block; each wave computes a 16 x (16*COLS) output strip:
// COLS independent WMMA accumulation chains share one A fragment per K-step.
// K/NN are compile-time so every stride is an immediate (no u64 muls in the
// loop). Requires M%16==0, NN%(16*COLS)==0, K%4==0. Full wave -> EXEC all-1s.
template <int COLS, int K, int NN>
__global__ void __launch_bounds__(32)
gemm_wmma_kernel(const float* __restrict__ A, const float* __restrict__ B,
                 float* __restrict__ C) {
    const int lane = threadIdx.x & 31;
    constexpr int TPR = NN / (16 * COLS);           // strips per 16-row band
    const int row0 = (blockIdx.x / TPR) << 4;
    const int col0 = (blockIdx.x % TPR) * (16 * COLS);

    const int mn = lane & 15;                       // M (for A) / N (for B)
    const int kk = (lane >> 4) << 1;                // K sub-offset: 0 or 2

    const float* Arow = A + (size_t)(row0 + mn) * K + kk;
    const float* Bp   = B + (size_t)kk * NN + col0 + mn;

    v8f acc[COLS];
#pragma unroll
    for (int c = 0; c < COLS; ++c) acc[c] = (v8f){};

#pragma unroll 4
    for (int k = 0; k < K; k += 4) {
        v2f a;
        a.x = Arow[k];
        a.y = Arow[k + 1];
        const float* B0 = Bp + (size_t)k * NN;
        v2f b[COLS];
#pragma unroll
        for (int c = 0; c < COLS; ++c) {
            b[c].x = B0[16 * c];
            b[c].y = B0[16 * c + NN];
        }
#pragma unroll
        for (int c = 0; c < COLS; ++c) {
            acc[c] = __builtin_amdgcn_wmma_f32_16x16x4_f32(
                false, a, false, b[c], (short)0, acc[c], false, false);
        }
    }

    // C/D layout: VGPR v -> M = v (lanes 0-15) or v+8 (lanes 16-31), N = lane&15
    const int mbase = row0 + ((lane >> 4) << 3);
    float* Cp = C + (size_t)mbase * NN + col0 + mn;
#pragma unroll
    for (int v = 0; v < 8; ++v) {
#pragma unroll
        for (int c = 0; c < COLS; ++c) {
            Cp[(size_t)v * NN + 16 * c] = acc[c][v];
        }
    }
}

// ------------------------------------------------------- attention logits
// as[n,h] = sum_c h[n,h,c]*a_src[h,c] ; ad likewise.
template <int H, int C>
__global__ void alpha_kernel(const float* __restrict__ h,
                             const float* __restrict__ a_s,
                             const float* __restrict__ a_d,
                             float* __restrict__ as, float* __restrict__ ad,
                             long long N) {
    long long gid = (long long)blockIdx.x * blockDim.x + threadIdx.x;
    if (gid >= N * H) return;
    int hh = (int)(gid % H);
    long long n = gid / H;
    const float* hp  = h + (size_t)n * H * C + (size_t)hh * C;
    const float* asp = a_s + (size_t)hh * C;
    const float* adp = a_d + (size_t)hh * C;
    float s = 0.f, d = 0.f;
#pragma unroll
    for (int c = 0; c < C; ++c) {
        float v = hp[c];
        s += v * asp[c];
        d += v * adp[c];
    }
    as[gid] = s;
    ad[gid] = d;
}

// ------------------------------------------------------------ edge helpers
__device__ __forceinline__ void edge_sd(const long long* __restrict__ src,
                                        const long long* __restrict__ dst,
                                        long long e, long long E,
                                        long long& s, long long& d) {
    if (e < E) { s = src[e]; d = dst[e]; }
    else       { s = d = e - E; }            // appended self-loop
}

__device__ __forceinline__ float leaky(float x) {
    return x > 0.f ? x : NEG_SLOPE * x;
}

// Monotone float atomic max via int/uint trick (init to -inf, no NaNs here).
__device__ __forceinline__ void atomic_max_f32(float* addr, float val) {
    if (val >= 0.f) atomicMax((int*)addr, __float_as_int(val));
    else            atomicMin((unsigned int*)addr, __float_as_uint(val));
}

// Pass 1: segment max of leaky_relu(as[src]+ad[dst]) grouped by dst.
template <int H>
__global__ void edge_max_kernel(const long long* __restrict__ src,
                                const long long* __restrict__ dst,
                                const float* __restrict__ as,
                                const float* __restrict__ ad,
                                float* __restrict__ m,
                                long long E, long long ET) {
    long long gid = (long long)blockIdx.x * blockDim.x + threadIdx.x;
    if (gid >= ET * H) return;
    int hh = (int)(gid % H);
    long long e = gid / H;
    long long s, d; edge_sd(src, dst, e, E, s, d);
    float ev = leaky(as[(size_t)s * H + hh] + ad[(size_t)d * H + hh]);
    atomic_max_f32(m + (size_t)d * H + hh, ev);
}

// Pass 2: w = exp(e - m[dst,h]); den[dst,h] += w; stash w in wbuf[e*H+h].
template <int H>
__global__ void edge_den_kernel(const long long* __restrict__ src,
                                const long long* __restrict__ dst,
                                const float* __restrict__ as,
                                const float* __restrict__ ad,
                                const float* __restrict__ m,
                                float* __restrict__ den,
                                float* __restrict__ wbuf,
                                long long E, long long ET) {
    long long gid = (long long)blockIdx.x * blockDim.x + threadIdx.x;
    if (gid >= ET * H) return;
    int hh = (int)(gid % H);
    long long e = gid / H;
    long long s, d; edge_sd(src, dst, e, E, s, d);
    float ev = leaky(as[(size_t)s * H + hh] + ad[(size_t)d * H + hh]);
    float w  = expf(ev - m[(size_t)d * H + hh]);
    wbuf[gid] = w;
    unsafeAtomicAdd(den + (size_t)d * H + hh, w);
}

// Pass 3: acc[dst,h,c] += h[src,h,c] * (wbuf[e,h] / den[dst,h]).
// One thread per (edge,h,c); lane index = c -> coalesced atomics; the
// per-(edge,h) scalars are broadcast loads that hit L2 (all live data < L2).
template <int H, int C>
__global__ void edge_scatter_kernel(const long long* __restrict__ src,
                                    const long long* __restrict__ dst,
                                    const float* __restrict__ h,
                                    const float* __restrict__ wbuf,
                                    const float* __restrict__ den,
                                    float* __restrict__ acc,
                                    long long E, long long ET) {
    long long gid = (long long)blockIdx.x * blockDim.x + threadIdx.x;
    if (gid >= ET * H * C) return;
    int c = (int)(gid % C);
    long long eh = gid / C;
    int hh = (int)(eh % H);
    long long e = eh / H;
    long long s, d; edge_sd(src, dst, e, E, s, d);
    float alpha = wbuf[eh] / den[(size_t)d * H + hh];
    float val = h[(size_t)s * H * C + (size_t)hh * C + c] * alpha;
    unsafeAtomicAdd(acc + (size_t)d * H * C + (size_t)hh * C + c, val);
}

// ---------------------------------------------------------------- launcher
extern "C" void kernel_launch(void* const* d_in, const int* in_sizes, int n_in,
                              void* d_out, int out_size, void* d_ws, size_t ws_size,
                              hipStream_t stream) {
    const float*     x     = (const float*)d_in[0];
    const long long* ei    = (const long long*)d_in[1];   // int64 [2, E]
    const float*     W1    = (const float*)d_in[2];
    const float*     asrc1 = (const float*)d_in[3];
    const float*     adst1 = (const float*)d_in[4];
    const float*     b1    = (const float*)d_in[5];
    const float*     W2    = (const float*)d_in[6];
    const float*     asrc2 = (const float*)d_in[7];
    const float*     adst2 = (const float*)d_in[8];
    const float*     b2    = (const float*)d_in[9];
    float*           out   = (float*)d_out;

    constexpr int FIN = 128, HC1 = 256, H1 = 8, C = 32;
    const long long N  = (long long)in_sizes[0] / FIN;   // 50000
    const long long E  = (long long)in_sizes[1] / 2;     // 800000
    const long long ET = E + N;                          // + self loops
    const long long* srcArr = ei;
    const long long* dstArr = ei + E;

    // workspace carve-up (floats)
    float* ws   = (float*)d_ws;
    float* h1   = ws;  ws += N * HC1;    // [N,8,32] layer-1 features
    float* acc1 = ws;  ws += N * HC1;    // layer-1 aggregate -> h2 after elu
    float* as1  = ws;  ws += N * H1;
    float* ad1  = ws;  ws += N * H1;
    float* m1   = ws;  ws += N * H1;
    float* den1 = ws;  ws += N * H1;
    float* g2   = ws;  ws += N * C;      // [N,1,32] layer-2 features
    float* as2  = ws;  ws += N;
    float* ad2  = ws;  ws += N;
    float* m2   = ws;  ws += N;
    float* den2 = ws;  ws += N;
    float* wbuf = ws;  ws += ET * H1;    // softmax numerators (reused layer 2)

    const int TB = 256;
    auto blocks = [](long long t, int b) { return (unsigned)((t + b - 1) / b); };

    // ---------------- layer 1 ----------------
    gemm_wmma_kernel<4, FIN, HC1>
        <<<(unsigned)((N / 16) * (HC1 / 64)), 32, 0, stream>>>(x, W1, h1);

    alpha_kernel<H1, C><<<blocks(N * H1, TB), TB, 0, stream>>>(
        h1, asrc1, adst1, as1, ad1, N);
    fill_kernel<<<blocks(N * H1, TB), TB, 0, stream>>>(m1, N * H1, -INFINITY);
    fill_kernel<<<blocks(N * H1, TB), TB, 0, stream>>>(den1, N * H1, 0.f);
    bias_init_kernel<HC1><<<blocks(N * HC1, TB), TB, 0, stream>>>(acc1, b1, N * HC1);

    edge_max_kernel<H1><<<blocks(ET * H1, TB), TB, 0, stream>>>(
        srcArr, dstArr, as1, ad1, m1, E, ET);
    edge_den_kernel<H1><<<blocks(ET * H1, TB), TB, 0, stream>>>(
        srcArr, dstArr, as1, ad1, m1, den1, wbuf, E, ET);
    edge_scatter_kernel<H1, C><<<blocks(ET * H1 * C, TB), TB, 0, stream>>>(
        srcArr, dstArr, h1, wbuf, den1, acc1, E, ET);

    elu_kernel<<<blocks(N * HC1, TB), TB, 0, stream>>>(acc1, N * HC1);  // acc1 -> h2

    // ---------------- layer 2 ----------------
    gemm_wmma_kernel<2, HC1, C>
        <<<(unsigned)((N / 16) * (C / 32)), 32, 0, stream>>>(acc1, W2, g2);

    alpha_kernel<1, C><<<blocks(N, TB), TB, 0, stream>>>(
        g2, asrc2, adst2, as2, ad2, N);
    fill_kernel<<<blocks(N, TB), TB, 0, stream>>>(m2, N, -INFINITY);
    fill_kernel<<<blocks(N, TB), TB, 0, stream>>>(den2, N, 0.f);
    bias_init_kernel<C><<<blocks(N * C, TB), TB, 0, stream>>>(out, b2, N * C);

    edge_max_kernel<1><<<blocks(ET, TB), TB, 0, stream>>>(
        srcArr, dstArr, as2, ad2, m2, E, ET);
    edge_den_kernel<1><<<blocks(ET, TB), TB, 0, stream>>>(
        srcArr, dstArr, as2, ad2, m2, den2, wbuf, E, ET);
    edge_scatter_kernel<1, C><<<blocks(ET * C, TB), TB, 0, stream>>>(
        srcArr, dstArr, g2, wbuf, den2, out, E, ET);
}